// SyntheticBlock_67989332296097
// MI455X (gfx1250) — compile-verified
//
#include <hip/hip_runtime.h>

#define N_NODES 100000
#define CH 64
#define NEDGE 1600000
#define SDIM 128
#define EPS_ 1e-5f
#define SLOPE 0.01f
#define EPW 16   // edges per wave in edge kernel

typedef __attribute__((ext_vector_type(2))) float v2f;
typedef __attribute__((ext_vector_type(8))) float v8f;

// ---------------------------------------------------------------------------
// CDNA5 f32 WMMA: D(16x16) = A(16x4) * B(4x16) + C, full fp32 precision.
// ---------------------------------------------------------------------------
static __device__ __forceinline__ v8f wmma4(v2f a, v2f b, v8f c) {
  return __builtin_amdgcn_wmma_f32_16x16x4_f32(false, a, false, b, (short)0, c,
                                               false, false);
}

// Load A-fragments for a 16xK tile from row-major src (row stride ld floats).
// f32 A layout (16x4 per chunk): lanes 0-15 rows M=0..15 hold K=+0,+1;
// lanes 16-31 hold K=+2,+3.
template <int KK>
static __device__ __forceinline__ void load_a(const float* src, int ld,
                                              v2f* afr, int lane) {
  int row = lane & 15;
  int kh = (lane >> 4) * 2;
  const float* p = src + row * ld + kh;
#pragma unroll
  for (int kk = 0; kk < KK; ++kk) {
    afr[kk].x = p[4 * kk];
    afr[kk].y = p[4 * kk + 1];
  }
}

// acc[t] (t=0..NT-1, 16-col tiles) += A(16 x 4*KK) @ W(4*KK x 16*NT),
// W row-major with row stride wld.
template <int KK, int NT>
static __device__ __forceinline__ void gemm_tiles(const v2f* afr,
                                                  const float* W, int wld,
                                                  v8f* acc, int lane) {
  int colb = lane & 15;
  int kh = (lane >> 4) * 2;
#pragma unroll
  for (int kk = 0; kk < KK; ++kk) {
    const float* wp = W + (4 * kk + kh) * wld + colb;
#pragma unroll
    for (int t = 0; t < NT; ++t) {
      v2f b;
      b.x = wp[16 * t];
      b.y = wp[wld + 16 * t];
      acc[t] = wmma4(afr[kk], b, acc[t]);
    }
  }
}

// ---------------------------------------------------------------------------
// zero agg
// ---------------------------------------------------------------------------
__global__ void zero_kernel(float4* __restrict__ p, int n4) {
  int i = blockIdx.x * blockDim.x + threadIdx.x;
  if (i < n4) p[i] = make_float4(0.f, 0.f, 0.f, 0.f);
}

// ---------------------------------------------------------------------------
// node_pre: per 16-node tile (one wave each):
//   h1    = relu(x @ h_w1 + h_b1)            (16x64, to LDS)
//   delta = tanh(h1 @ h_w2 + h_b2)           (16x3, to global)
//   xf    = x @ f_w[3:,:] + f_b              (16x64, to global; relu deferred)
// ---------------------------------------------------------------------------
__global__ __launch_bounds__(256) void node_pre_kernel(
    const float* __restrict__ x, const float* __restrict__ h_w1,
    const float* __restrict__ h_b1, const float* __restrict__ h_w2,
    const float* __restrict__ h_b2, const float* __restrict__ f_w,
    const float* __restrict__ f_b, float* __restrict__ xf,
    float* __restrict__ delta) {
  __shared__ float lds[8][16 * 65];
  int wave = threadIdx.x >> 5;
  int lane = threadIdx.x & 31;
  int tile = blockIdx.x * 8 + wave;
  if (tile >= N_NODES / 16) return;
  int base = tile * 16;
  float* tl = lds[wave];

  int colb = lane & 15;
  int mhi = (lane >> 4) * 8;

  v2f afr[16];
  load_a<16>(x + (size_t)base * CH, CH, afr, lane);

  // GEMM1: h1 = relu(x @ h_w1 + h_b1) -> LDS
  v8f acc[4] = {};
  gemm_tiles<16, 4>(afr, h_w1, CH, acc, lane);
#pragma unroll
  for (int t = 0; t < 4; ++t) {
    float b = h_b1[16 * t + colb];
#pragma unroll
    for (int r = 0; r < 8; ++r) {
      float v = acc[t][r] + b;
      v = fmaxf(v, 0.f);
      tl[(r + mhi) * 65 + 16 * t + colb] = v;
    }
  }

  // GEMM2: xf = x @ f_w[3:,:] + f_b -> global (A-fragments reused)
  v8f acc2[4] = {};
  gemm_tiles<16, 4>(afr, f_w + 3 * CH, CH, acc2, lane);
#pragma unroll
  for (int t = 0; t < 4; ++t) {
    float b = f_b[16 * t + colb];
#pragma unroll
    for (int r = 0; r < 8; ++r) {
      xf[(size_t)(base + r + mhi) * CH + 16 * t + colb] = acc2[t][r] + b;
    }
  }

  __builtin_amdgcn_wave_barrier();  // order LDS h1 writes before reads

  // delta head: 64 -> 3, tanh (lanes 0..15 each own one node row)
  if (lane < 16) {
    const float* hr = tl + lane * 65;
    float d0 = h_b2[0], d1 = h_b2[1], d2 = h_b2[2];
#pragma unroll 8
    for (int k = 0; k < CH; ++k) {
      float hv = hr[k];
      d0 = fmaf(hv, h_w2[k * 3 + 0], d0);
      d1 = fmaf(hv, h_w2[k * 3 + 1], d1);
      d2 = fmaf(hv, h_w2[k * 3 + 2], d2);
    }
    int n = base + lane;
    delta[n * 3 + 0] = tanhf(d0);
    delta[n * 3 + 1] = tanhf(d1);
    delta[n * 3 + 2] = tanhf(d2);
  }
}

// ---------------------------------------------------------------------------
// edge kernel: one wave per edge, EPW edges per wave.
//   rel = pos[s]-pos[d]+delta[d]
//   m   = relu(xf[s] + rel0*W0 + rel1*W1 + rel2*W2)   (64 ch, 2/lane)
//   agg[d] += m  (hardware f32 atomics)
// ---------------------------------------------------------------------------
__global__ __launch_bounds__(256) void edge_kernel(
    const float* __restrict__ xf, const float* __restrict__ pos,
    const float* __restrict__ delta, const int* __restrict__ ei,
    const float* __restrict__ fw, float* __restrict__ agg) {
  int gwave = (blockIdx.x * blockDim.x + threadIdx.x) >> 5;
  int lane = threadIdx.x & 31;
  int c0 = lane * 2;

  const float2 w0 = *(const float2*)(fw + 0 * CH + c0);
  const float2 w1 = *(const float2*)(fw + 1 * CH + c0);
  const float2 w2 = *(const float2*)(fw + 2 * CH + c0);

  int e0 = gwave * EPW;
#pragma unroll 1
  for (int i = 0; i < EPW; ++i) {
    int e = e0 + i;
    if (e >= NEDGE) return;
    int s = ei[e];
    int d = ei[NEDGE + e];

    float rel = 0.f;
    if (lane < 3)
      rel = pos[s * 3 + lane] - pos[d * 3 + lane] + delta[d * 3 + lane];
    float r0 = __shfl(rel, 0, 32);
    float r1 = __shfl(rel, 1, 32);
    float r2 = __shfl(rel, 2, 32);

    float2 xv = *(const float2*)(xf + (size_t)s * CH + c0);
    float m0 = fmaf(r0, w0.x, fmaf(r1, w1.x, fmaf(r2, w2.x, xv.x)));
    float m1 = fmaf(r0, w0.y, fmaf(r1, w1.y, fmaf(r2, w2.y, xv.y)));
    m0 = fmaxf(m0, 0.f);
    m1 = fmaxf(m1, 0.f);

    float* dst = agg + (size_t)d * CH + c0;
    unsafeAtomicAdd(dst, m0);
    unsafeAtomicAdd(dst + 1, m1);
  }
}

// ---------------------------------------------------------------------------
// node_post: per 16-node tile:
//   y  = relu(relu(agg@g_w1+g_b1)@g_w2+g_b2)
//   s  = style@s_w + s_b ;  gamma=s[:, :64], beta=s[:,64:]
//   o  = leaky( gamma*(y-mu)/sqrt(var+eps) + beta )
// ---------------------------------------------------------------------------
__global__ __launch_bounds__(256) void node_post_kernel(
    const float* __restrict__ agg, const float* __restrict__ style,
    const float* __restrict__ g_w1, const float* __restrict__ g_b1,
    const float* __restrict__ g_w2, const float* __restrict__ g_b2,
    const float* __restrict__ s_w, const float* __restrict__ s_b,
    float* __restrict__ out) {
  __shared__ float lds[8][16 * 65 + 32];
  int wave = threadIdx.x >> 5;
  int lane = threadIdx.x & 31;
  int tile = blockIdx.x * 8 + wave;
  if (tile >= N_NODES / 16) return;
  int base = tile * 16;
  float* tl = lds[wave];
  float* stats = tl + 16 * 65;

  int colb = lane & 15;
  int mhi = (lane >> 4) * 8;

  // t1 = relu(agg @ g_w1 + g_b1) -> LDS
  v2f afr[16];
  load_a<16>(agg + (size_t)base * CH, CH, afr, lane);
  v8f acc[4] = {};
  gemm_tiles<16, 4>(afr, g_w1, CH, acc, lane);
#pragma unroll
  for (int t = 0; t < 4; ++t) {
    float b = g_b1[16 * t + colb];
#pragma unroll
    for (int r = 0; r < 8; ++r)
      tl[(r + mhi) * 65 + 16 * t + colb] = fmaxf(acc[t][r] + b, 0.f);
  }
  __builtin_amdgcn_wave_barrier();

  // y = relu(t1 @ g_w2 + g_b2) -> regs + LDS (overwrite t1)
  v2f afr2[16];
  load_a<16>(tl, 65, afr2, lane);
  v8f accy[4] = {};
  gemm_tiles<16, 4>(afr2, g_w2, CH, accy, lane);
#pragma unroll
  for (int t = 0; t < 4; ++t) {
    float b = g_b2[16 * t + colb];
#pragma unroll
    for (int r = 0; r < 8; ++r) {
      float v = fmaxf(accy[t][r] + b, 0.f);
      accy[t][r] = v;
      tl[(r + mhi) * 65 + 16 * t + colb] = v;
    }
  }
  __builtin_amdgcn_wave_barrier();

  // per-row mean / rsigma (lanes 0..15 each own one node row)
  if (lane < 16) {
    const float* yr = tl + lane * 65;
    float mu = 0.f;
#pragma unroll 8
    for (int k = 0; k < CH; ++k) mu += yr[k];
    mu *= (1.f / CH);
    float var = 0.f;
#pragma unroll 8
    for (int k = 0; k < CH; ++k) {
      float dv = yr[k] - mu;
      var = fmaf(dv, dv, var);
    }
    var *= (1.f / CH);
    stats[lane] = mu;
    stats[16 + lane] = rsqrtf(var + EPS_);
  }
  __builtin_amdgcn_wave_barrier();

  // style GEMM: (16x128) @ (128x128) -> gamma (tiles 0..3), beta (tiles 4..7)
  v2f afs[32];
  load_a<32>(style + (size_t)base * SDIM, SDIM, afs, lane);
  v8f accs[8] = {};
  gemm_tiles<32, 8>(afs, s_w, SDIM, accs, lane);

  // AdaIN + leaky relu, store
#pragma unroll
  for (int t = 0; t < 4; ++t) {
    float gb = s_b[16 * t + colb];
    float bb = s_b[CH + 16 * t + colb];
#pragma unroll
    for (int r = 0; r < 8; ++r) {
      int m = r + mhi;
      float mu = stats[m];
      float rs = stats[16 + m];
      float g = accs[t][r] + gb;
      float bt = accs[t + 4][r] + bb;
      float o = fmaf(g, (accy[t][r] - mu) * rs, bt);
      o = (o >= 0.f) ? o : SLOPE * o;
      out[(size_t)(base + m) * CH + 16 * t + colb] = o;
    }
  }
}

// ---------------------------------------------------------------------------
// launch
// ---------------------------------------------------------------------------
extern "C" void kernel_launch(void* const* d_in, const int* in_sizes, int n_in,
                              void* d_out, int out_size, void* d_ws,
                              size_t ws_size, hipStream_t stream) {
  (void)in_sizes; (void)n_in; (void)out_size; (void)ws_size;
  const float* h = (const float*)d_in[0];
  const float* pos = (const float*)d_in[1];
  const int* ei = (const int*)d_in[2];
  const float* style = (const float*)d_in[3];
  const float* p1[12];
  const float* p2[12];
  for (int i = 0; i < 12; ++i) {
    p1[i] = (const float*)d_in[4 + i];
    p2[i] = (const float*)d_in[16 + i];
  }
  // param order: 0 h_w1, 1 h_b1, 2 h_w2, 3 h_b2, 4 f_w, 5 f_b,
  //              6 g_w1, 7 g_b1, 8 g_w2, 9 g_b2, 10 s_w, 11 s_b

  float* xf = (float*)d_ws;
  float* agg = xf + (size_t)N_NODES * CH;
  float* dl = agg + (size_t)N_NODES * CH;
  float* hmid = dl + (size_t)N_NODES * 3;

  int ntileb = (N_NODES / 16 + 7) / 8;                 // 782
  int edgeb = ((NEDGE + EPW - 1) / EPW + 7) / 8;       // 12500 (8 waves/block)
  int zb = (N_NODES * CH / 4 + 255) / 256;             // 6250

  for (int blk = 0; blk < 2; ++blk) {
    const float* const* P = blk ? p2 : p1;
    const float* xin = blk ? hmid : h;
    float* xout = blk ? (float*)d_out : hmid;

    zero_kernel<<<zb, 256, 0, stream>>>((float4*)agg, N_NODES * CH / 4);
    node_pre_kernel<<<ntileb, 256, 0, stream>>>(xin, P[0], P[1], P[2], P[3],
                                                P[4], P[5], xf, dl);
    edge_kernel<<<edgeb, 256, 0, stream>>>(xf, pos, dl, ei, P[4], agg);
    node_post_kernel<<<ntileb, 256, 0, stream>>>(agg, style, P[6], P[7], P[8],
                                                 P[9], P[10], P[11], xout);
  }
}